// MultiBoxHeadLoss_90091234000967
// MI455X (gfx1250) — compile-verified
//
#include <hip/hip_runtime.h>
#include <hip/hip_bf16.h>
#include <math.h>
#include <stdint.h>

typedef __attribute__((ext_vector_type(2))) float v2f;
typedef __attribute__((ext_vector_type(8))) float v8f;

#define THRESH 0.5f

// Wave-level f32 sum using the CDNA5 f32 WMMA as a reduction engine.
// A = ones(16x4), B holds one value per lane (other slot 0):
// D[m][n] = sum_k B[k][n]  (all rows identical). Each of the 64 B slots maps
// to exactly one (k,n), so sum over the 16 columns = wave sum. c[0] across all
// 32 lanes covers the 16 column sums twice (rows M=0 and M=8), hence *0.5.
__device__ __forceinline__ float wave_sum_wmma(float v) {
  v2f a; a.x = 1.0f; a.y = 1.0f;
  v2f b; b.x = v;    b.y = 0.0f;
  v8f c = {};
  c = __builtin_amdgcn_wmma_f32_16x16x4_f32(false, a, false, b, (short)0, c,
                                            false, false);
  float s = c[0];
#pragma unroll
  for (int off = 16; off > 0; off >>= 1) s += __shfl_xor(s, off, 32);
  return 0.5f * s;
}

// ---------------- kernel 0: zero accumulators (must run every launch) -------
__global__ void k_init(unsigned long long* gbest, int nbest, int* num_pos,
                       int B, float* scalars, float* loss_c_neg) {
  int t = blockIdx.x * blockDim.x + threadIdx.x;
  for (int i = t; i < nbest; i += gridDim.x * blockDim.x) gbest[i] = 0ull;
  if (t < B) { num_pos[t] = 0; loss_c_neg[t] = 0.f; }
  if (t < 2) scalars[t] = 0.f;
}

// ---------------- kernel 1: IoU matching ------------------------------------
// grid (P/256, B), block 256. Stages the batch's 32 truths into LDS with the
// gfx1250 async-to-LDS path, computes per-prior best truth (max/argmax over
// axis 0) and per-truth best prior (argmax over axis 1) via packed u64 max.
__global__ __launch_bounds__(256) void k_match(
    const float* __restrict__ priors, const float* __restrict__ targets,
    float* __restrict__ bto, int* __restrict__ bti,
    unsigned long long* __restrict__ gbest, int P, int O) {
  __shared__ float sT[160];                 // 32 truths x 5 floats
  __shared__ unsigned long long sBest[32];  // per-truth packed best prior
  const int b = blockIdx.y;
  const int tid = threadIdx.x;
  const int p = blockIdx.x * 256 + tid;

  if (tid < 32) {                           // wave 0 only, full EXEC
    sBest[tid] = 0ull;
    const float* gsrc = targets + (size_t)b * O * 5;
#pragma unroll
    for (int i = 0; i < 5; ++i) {
      int idx = i * 32 + tid;
      unsigned ldsoff = (unsigned)(unsigned long long)&sT[idx];
      const float* g = gsrc + idx;
      asm volatile("global_load_async_to_lds_b32 %0, %1, off"
                   :: "v"(ldsoff), "v"(g) : "memory");
    }
    asm volatile("s_wait_asynccnt 0x0" ::: "memory");
  }
  __syncthreads();

  float4 pr = reinterpret_cast<const float4*>(priors)[p];  // cx cy w h
  float px1 = pr.x - 0.5f * pr.z, py1 = pr.y - 0.5f * pr.w;
  float px2 = pr.x + 0.5f * pr.z, py2 = pr.y + 0.5f * pr.w;
  float areaB = (px2 - px1) * (py2 - py1);

  float bov = -1.0f; int bidx = 0;
  unsigned packlo = 0xFFFFFFFFu - (unsigned)p;  // first-index tie-break
#pragma unroll 4
  for (int t = 0; t < 32; ++t) {
    float ax1 = sT[t*5+0], ay1 = sT[t*5+1], ax2 = sT[t*5+2], ay2 = sT[t*5+3];
    float iw = fminf(ax2, px2) - fmaxf(ax1, px1);
    float ih = fminf(ay2, py2) - fmaxf(ay1, py1);
    iw = fmaxf(iw, 0.f); ih = fmaxf(ih, 0.f);
    float inter = iw * ih;
    float areaA = (ax2 - ax1) * (ay2 - ay1);
    float iou = inter / (areaA + areaB - inter);
    if (iou > bov) { bov = iou; bidx = t; }  // first-occurrence argmax
    unsigned long long packed =
        ((unsigned long long)__float_as_uint(iou) << 32) | packlo;
    atomicMax(&sBest[t], packed);            // ds_max_u64
  }
  bto[(size_t)b * P + p] = bov;
  bti[(size_t)b * P + p] = bidx;
  __syncthreads();
  if (tid < 32) atomicMax(&gbest[b * 32 + tid], sBest[tid]);
}

// ---------------- kernel 2: forced matches (ordered, last write wins) -------
__global__ void k_force(const unsigned long long* __restrict__ gbest,
                        float* __restrict__ bto, int* __restrict__ bti,
                        int P, int O, int B) {
  int b = threadIdx.x;
  if (b >= B) return;
  for (int o = 0; o < O; ++o) {            // sequential: JAX scatter order
    unsigned long long v = gbest[b * O + o];
    int p = (int)(0xFFFFFFFFu - (unsigned)(v & 0xFFFFFFFFull));
    bto[(size_t)b * P + p] = 2.0f;
    bti[(size_t)b * P + p] = o;
  }
}

// ---------------- kernel 3: per-prior losses + WMMA reductions --------------
__global__ __launch_bounds__(256) void k_loss(
    const float* __restrict__ loc_data, const float* __restrict__ conf_data,
    const float* __restrict__ priors, const float* __restrict__ targets,
    const float* __restrict__ bto, const int* __restrict__ bti,
    float* __restrict__ ce_mine, float* __restrict__ scalars,
    int* __restrict__ num_pos, int P, int O) {
  const int b = blockIdx.y;
  const int tid = threadIdx.x;
  const int p = blockIdx.x * 256 + tid;
  const size_t idx = (size_t)b * P + p;

  float ov = bto[idx];
  int ti = bti[idx];
  const float* T = targets + ((size_t)b * O + ti) * 5;
  float tx1 = T[0], ty1 = T[1], tx2 = T[2], ty2 = T[3];
  int conf = (ov < THRESH) ? 0 : (int)T[4];
  bool pos = conf > 0;

  float4 pr = reinterpret_cast<const float4*>(priors)[p];
  float gcx = ((tx1 + tx2) * 0.5f - pr.x) / (0.1f * pr.z);
  float gcy = ((ty1 + ty2) * 0.5f - pr.y) / (0.1f * pr.w);
  float gw = logf((tx2 - tx1) / pr.z) / 0.2f;
  float gh = logf((ty2 - ty1) / pr.w) / 0.2f;

  const float* L = loc_data + idx * 3;
  float q0 = L[0], q1 = L[1], q2 = L[2];
  float d0 = q0 - gcx, d1 = q1 - gcy, d2 = q2 - gw, d3 = q2 - gh;
  float a0 = fabsf(d0), a1 = fabsf(d1), a2 = fabsf(d2), a3 = fabsf(d3);
  float lsum = (a0 < 1.f ? 0.5f * d0 * d0 : a0 - 0.5f) +
               (a1 < 1.f ? 0.5f * d1 * d1 : a1 - 0.5f) +
               (a2 < 1.f ? 0.5f * d2 * d2 : a2 - 0.5f) +
               (a3 < 1.f ? 0.5f * d3 * d3 : a3 - 0.5f);
  float ll = pos ? lsum : 0.f;

  const float* C = conf_data + idx * 2;
  float c0 = C[0], c1 = C[1];
  float mx = fmaxf(c0, c1);
  float lse = mx + logf(expf(c0 - mx) + expf(c1 - mx));
  float ce = lse - ((conf == 0) ? c0 : c1);
  ce_mine[idx] = pos ? 0.f : ce;
  float lcp = pos ? ce : 0.f;

  float wl = wave_sum_wmma(ll);     // v_wmma_f32_16x16x4_f32
  float wc = wave_sum_wmma(lcp);
  int wn = __popcll(__ballot(pos));

  __shared__ float sL, sC;
  __shared__ int sN;
  if (tid == 0) { sL = 0.f; sC = 0.f; sN = 0; }
  __syncthreads();
  if ((tid & 31) == 0) { atomicAdd(&sL, wl); atomicAdd(&sC, wc); atomicAdd(&sN, wn); }
  __syncthreads();
  if (tid == 0) {
    atomicAdd(&scalars[0], sL);
    atomicAdd(&scalars[1], sC);
    atomicAdd(&num_pos[b], sN);
  }
}

// ---------------- kernel 4: exact top-k sum via 32-bit radix select ---------
// One block per batch. ce_mine >= 0 so float bits are order-isomorphic.
// Sum of top-k = sum(x > v*) + remaining * v*, exact under stable-sort ties.
__global__ __launch_bounds__(256) void k_select(
    const float* __restrict__ ce_mine, const int* __restrict__ num_pos,
    float* __restrict__ loss_c_neg, int P) {
  const int b = blockIdx.x;
  const int tid = threadIdx.x;
  const float* x = ce_mine + (size_t)b * P;

  int np = num_pos[b];
  long long k64 = 3LL * np;
  if (k64 > P - 1) k64 = P - 1;
  int k = (int)k64;
  if (k <= 0) { if (tid == 0) loss_c_neg[b] = 0.f; return; }

  __shared__ unsigned hist[256];
  __shared__ unsigned s_sel, s_kk;
  unsigned prefix = 0, kk = (unsigned)k;

  for (int shift = 24; shift >= 0; shift -= 8) {
    hist[tid] = 0;
    __syncthreads();
    for (int i = tid; i < P; i += 256) {
      unsigned key = __float_as_uint(x[i]);
      bool m = (shift == 24) || ((key >> (shift + 8)) == prefix);
      if (m) atomicAdd(&hist[(key >> shift) & 0xFFu], 1u);
    }
    __syncthreads();
    if (tid == 0) {
      unsigned cum = 0, sel = 0, rem = kk;
      for (int bin = 255; bin >= 0; --bin) {
        unsigned cnt = hist[bin];
        if (cum + cnt >= kk) { sel = (unsigned)bin; rem = kk - cum; break; }
        cum += cnt;
      }
      s_sel = sel; s_kk = rem;
    }
    __syncthreads();
    prefix = (prefix << 8) | s_sel;
    kk = s_kk;
    __syncthreads();
  }

  float vstar = __uint_as_float(prefix);
  float ssum = 0.f;
  for (int i = tid; i < P; i += 256) {
    unsigned key = __float_as_uint(x[i]);
    if (key > prefix) ssum += x[i];
  }
  float ws = wave_sum_wmma(ssum);   // v_wmma_f32_16x16x4_f32
  __shared__ float sPart;
  if (tid == 0) sPart = 0.f;
  __syncthreads();
  if ((tid & 31) == 0) atomicAdd(&sPart, ws);
  __syncthreads();
  if (tid == 0) loss_c_neg[b] = sPart + (float)kk * vstar;
}

// ---------------- kernel 5: finalize ----------------------------------------
__global__ void k_final(const float* __restrict__ scalars,
                        const int* __restrict__ num_pos,
                        const float* __restrict__ loss_c_neg,
                        float* __restrict__ out, int B) {
  if (threadIdx.x == 0 && blockIdx.x == 0) {
    int N = 0; float neg = 0.f;
    for (int i = 0; i < B; ++i) { N += num_pos[i]; neg += loss_c_neg[i]; }
    float fN = (float)N;
    out[0] = scalars[0] / fN;
    out[1] = (scalars[1] + neg) / fN;
  }
}

extern "C" void kernel_launch(void* const* d_in, const int* in_sizes, int n_in,
                              void* d_out, int out_size, void* d_ws,
                              size_t ws_size, hipStream_t stream) {
  const float* loc_data  = (const float*)d_in[0];
  const float* conf_data = (const float*)d_in[1];
  const float* priors    = (const float*)d_in[2];
  const float* targets   = (const float*)d_in[3];
  const int P = in_sizes[2] / 4;            // 32768
  const int B = in_sizes[0] / (P * 3);      // 32
  const int O = in_sizes[3] / (B * 5);      // 32
  float* out = (float*)d_out;

  char* w = (char*)d_ws;
  size_t off = 0;
  float* bto = (float*)(w + off); off += (size_t)B * P * 4;
  int*   bti = (int*)(w + off);   off += (size_t)B * P * 4;
  float* cem = (float*)(w + off); off += (size_t)B * P * 4;
  unsigned long long* gbest = (unsigned long long*)(w + off); off += (size_t)B * O * 8;
  int*   num_pos = (int*)(w + off); off += (size_t)B * 4;
  float* scalars = (float*)(w + off); off += 8;
  float* lcn = (float*)(w + off); off += (size_t)B * 4;

  k_init<<<4, 256, 0, stream>>>(gbest, B * O, num_pos, B, scalars, lcn);
  dim3 g1(P / 256, B);
  k_match<<<g1, 256, 0, stream>>>(priors, targets, bto, bti, gbest, P, O);
  k_force<<<1, B, 0, stream>>>(gbest, bto, bti, P, O, B);
  k_loss<<<g1, 256, 0, stream>>>(loc_data, conf_data, priors, targets, bto,
                                 bti, cem, scalars, num_pos, P, O);
  k_select<<<B, 256, 0, stream>>>(cem, num_pos, lcn, P);
  k_final<<<1, 32, 0, stream>>>(scalars, num_pos, lcn, out, B);
}